// EffSparseEdgeOnlyConv_79199196938681
// MI455X (gfx1250) — compile-verified
//
#include <hip/hip_runtime.h>
#include <hip/hip_bf16.h>

// ---------------------------------------------------------------------------
// out = A@x2 - x2*deg  with  x2 = x@W + b
//     = per node i: sum over edges e with dst(e)=i of (x2[src_e] - x2[i])
// Kernel 1: zero d_out
// Kernel 2: x2 = x@W + b via V_WMMA_F32_16X16X4_F32 (full fp32 precision)
// Kernel 3: per-edge fused scatter: atomicAdd(out[dst], x2[src]-x2[dst])
// ---------------------------------------------------------------------------

typedef __attribute__((ext_vector_type(2))) float v2f;
typedef __attribute__((ext_vector_type(8))) float v8f;

#define C_DIM 128  // C_IN == C_OUT == 128

// ---------------------------------------------------------------- zero out
__global__ void zero_f4_kernel(float4* __restrict__ p, int n4) {
    int i = blockIdx.x * blockDim.x + threadIdx.x;
    if (i < n4) {
        float4 z; z.x = 0.f; z.y = 0.f; z.z = 0.f; z.w = 0.f;
        p[i] = z;
    }
}

// ------------------------------------------------- x2 = x @ W + b  (WMMA f32)
// Block: 256 threads = 8 waves. Block b -> rows [16b, 16b+16).
// Wave w -> columns [16w, 16w+16). K loop: 128/4 = 32 WMMA_F32_16X16X4_F32.
//
// A (16x4 f32) lane layout: lanes 0-15 hold M=lane, K={0,1}; lanes 16-31 hold
// M=lane-16, K={2,3}  -> one contiguous v2f load per lane.
// B (4x16 f32) mirrored: lanes 0-15 hold N=lane, K={0,1}; lanes 16-31 K={2,3}.
// C/D (16x16 f32, 8 VGPRs): VGPR v, lanes 0-15 -> M=v, lanes 16-31 -> M=v+8.
__global__ void __launch_bounds__(256)
gemm_bias_wmma_kernel(const float* __restrict__ x,
                      const float* __restrict__ W,
                      const float* __restrict__ bias,
                      float* __restrict__ x2,
                      int M) {
    const int lane = threadIdx.x & 31;
    const int wave = threadIdx.x >> 5;     // 0..7 -> N tile
    const int l    = lane & 15;
    const int half = lane >> 4;            // 0: K=0,1 / rows v ; 1: K=2,3 / rows v+8

    const int m0 = blockIdx.x * 16;
    const int n0 = wave * 16;
    if (m0 >= M) return;

    const float* arow = x + (size_t)(m0 + l) * C_DIM;   // A row for this lane
    const int    col  = n0 + l;                         // B/C column for this lane

    v8f acc = {};
#pragma unroll 4
    for (int k = 0; k < C_DIM; k += 4) {
        const int kk = k + 2 * half;
        v2f a = *(const v2f*)(arow + kk);               // x[m0+l][kk], x[m0+l][kk+1]
        v2f bm;
        bm.x = W[(size_t)kk * C_DIM + col];             // W[kk]  [col]
        bm.y = W[(size_t)(kk + 1) * C_DIM + col];       // W[kk+1][col]
        // D = A(16x4) * B(4x16) + C   (emits v_wmma_f32_16x16x4_f32)
        acc = __builtin_amdgcn_wmma_f32_16x16x4_f32(
            /*neg_a=*/false, a, /*neg_b=*/false, bm,
            /*c_mod=*/(short)0, acc, /*reuse_a=*/false, /*reuse_b=*/false);
    }

    const float bv = bias[col];
    const int rowbase = m0 + half * 8;
#pragma unroll
    for (int v = 0; v < 8; ++v) {
        x2[(size_t)(rowbase + v) * C_DIM + col] = acc[v] + bv;
    }
}

// ------------------------------------------- fused edge scatter (one wave/edge)
// lane handles 4 contiguous channels (128 ch / 32 lanes). Gathers are
// coalesced 512B bursts; x2 and out are L2-resident (51.2MB each << 192MB L2).
// Hardware f32 atomics via relaxed/agent scope -> global_atomic_add_f32.
__global__ void __launch_bounds__(256)
edge_scatter_kernel(const long long* __restrict__ ei,   // [2, E] int64
                    const float* __restrict__ x2,
                    float* __restrict__ out,
                    int n_edges) {
    const int e = blockIdx.x * 8 + (int)(threadIdx.x >> 5);
    if (e >= n_edges) return;
    const int lane = threadIdx.x & 31;

    const long long dst = ei[e];              // aggregation target
    const long long src = ei[(size_t)n_edges + e];  // message source

    const float4 s = *((const float4*)(x2 + (size_t)src * C_DIM) + lane);
    const float4 d = *((const float4*)(x2 + (size_t)dst * C_DIM) + lane);

    float* o = out + (size_t)dst * C_DIM + (size_t)lane * 4;
    __hip_atomic_fetch_add(o + 0, s.x - d.x, __ATOMIC_RELAXED, __HIP_MEMORY_SCOPE_AGENT);
    __hip_atomic_fetch_add(o + 1, s.y - d.y, __ATOMIC_RELAXED, __HIP_MEMORY_SCOPE_AGENT);
    __hip_atomic_fetch_add(o + 2, s.z - d.z, __ATOMIC_RELAXED, __HIP_MEMORY_SCOPE_AGENT);
    __hip_atomic_fetch_add(o + 3, s.w - d.w, __ATOMIC_RELAXED, __HIP_MEMORY_SCOPE_AGENT);
}

// ---------------------------------------------------------------------------
extern "C" void kernel_launch(void* const* d_in, const int* in_sizes, int n_in,
                              void* d_out, int out_size, void* d_ws, size_t ws_size,
                              hipStream_t stream) {
    const float*     x  = (const float*)d_in[0];       // [N, 128] fp32
    const long long* ei = (const long long*)d_in[1];   // [2, E] int64
    const float*     W  = (const float*)d_in[2];       // [128, 128] fp32
    const float*     b  = (const float*)d_in[3];       // [128] fp32
    float* out = (float*)d_out;                        // [N, 128] fp32
    float* x2  = (float*)d_ws;                         // scratch: [N, 128] fp32

    const int M       = in_sizes[0] / C_DIM;           // 100000 nodes
    const int n_edges = in_sizes[1] / 2;               // 1.6M edges
    (void)n_in; (void)out_size; (void)ws_size;

    // 1) out = 0
    {
        const int n4 = (M * C_DIM) / 4;
        zero_f4_kernel<<<(n4 + 255) / 256, 256, 0, stream>>>((float4*)out, n4);
    }
    // 2) x2 = x @ W + b   (M is a multiple of 16: 100000 = 16 * 6250)
    {
        const int mtiles = (M + 15) / 16;
        gemm_bias_wmma_kernel<<<mtiles, 256, 0, stream>>>(x, W, b, x2, M);
    }
    // 3) out[dst] += x2[src] - x2[dst]  over all edges (8 edges per 256-block)
    {
        const int blocks = (n_edges + 7) / 8;
        edge_scatter_kernel<<<blocks, 256, 0, stream>>>(ei, x2, out, n_edges);
    }
}